// EnhancedCorrelationGNN_47605417508977
// MI455X (gfx1250) — compile-verified
//
#include <hip/hip_runtime.h>
#include <hip/hip_bf16.h>
#include <math.h>

typedef __attribute__((ext_vector_type(2))) float v2f;
typedef __attribute__((ext_vector_type(8))) float v8f;

#define N_NODES 50000
#define N_EDGES 800000
#define IN_F    128
#define N_HEADS 8
#define HEAD_D  16
#define OUT_F   128
#define LRELU_A 0.2f

// Workspace layout (in floats)
#define WS_H     0          // N*128  projected features
#define WS_AS    6400000    // N*8    attn_src logits
#define WS_AD    6800000    // N*8    attn_dst logits
#define WS_MKEY  7200000    // N*8    segment-max keys (uint)
#define WS_SSUM  7600000    // N*8    segment exp-sums
#define WS_EBUF  8000000    // E*8    edge scores -> exp values
// total 14,400,000 floats = 57.6 MB

// Monotonic float<->uint key for atomic max (0 == "-inf" sentinel)
__device__ __forceinline__ unsigned fkey(float f) {
    unsigned u = __float_as_uint(f);
    return (u & 0x80000000u) ? ~u : (u | 0x80000000u);
}
__device__ __forceinline__ float fdecode(unsigned k) {
    return (k & 0x80000000u) ? __uint_as_float(k ^ 0x80000000u)
                             : __uint_as_float(~k);
}

// ---------------------------------------------------------------------------
// Kernel 1: h[N,128] = x[N,128] @ B[128,128], B[k][h*16+d] = W[h][k][d]
// One block = 8 waves; wave w computes head w's 16x16 tile for a 16-node tile.
// Uses V_WMMA_F32_16X16X4_F32 (f32, matches reference precision).
// ---------------------------------------------------------------------------
__global__ void __launch_bounds__(256)
gat_gemm_wmma(const float* __restrict__ x, const float* __restrict__ W,
              float* __restrict__ hout) {
    __shared__ float sX[16 * 132];   // 16 rows x 128 cols, padded (bank-conflict free)

    const int t    = threadIdx.x;
    const int wave = t >> 5;         // 0..7 == head
    const int lane = t & 31;
    const int half = lane >> 4;      // 0: lanes 0-15, 1: lanes 16-31
    const int l    = lane & 15;

    const float* Wh = W + wave * (IN_F * HEAD_D);   // this head's 128x16 block
    const int numTiles = N_NODES / 16;              // 3125 exactly

    for (int mt = blockIdx.x; mt < numTiles; mt += gridDim.x) {
        const int m0 = mt * 16;

        __syncthreads();
        // cooperative load of x tile: 16x128 floats, fully coalesced
        for (int i = t; i < 16 * IN_F; i += 256)
            sX[(i >> 7) * 132 + (i & 127)] = x[m0 * IN_F + i];
        __syncthreads();

        // prefetch next tile for this block (speculative, gfx1250 global_prefetch)
        {
            int nmt = mt + gridDim.x;
            if (nmt < numTiles)
                __builtin_prefetch(&x[nmt * 16 * IN_F], 0, 0);
        }

        v8f c = {0.f, 0.f, 0.f, 0.f, 0.f, 0.f, 0.f, 0.f};
        #pragma unroll
        for (int kk = 0; kk < IN_F; kk += 4) {
            // A 16x4: VGPR0 = K = kk+2*half, VGPR1 = K = kk+2*half+1, row M = l
            v2f a, b;
            a.x = sX[l * 132 + kk + 2 * half];
            a.y = sX[l * 132 + kk + 2 * half + 1];
            // B 4x16: same K split across halves, column N = l
            b.x = Wh[(kk + 2 * half)     * HEAD_D + l];
            b.y = Wh[(kk + 2 * half + 1) * HEAD_D + l];
            c = __builtin_amdgcn_wmma_f32_16x16x4_f32(
                    false, a, false, b, (short)0, c, false, false);
        }

        // C/D layout: VGPR r = row (r + 8*half), col = l
        #pragma unroll
        for (int r = 0; r < 8; ++r)
            hout[(m0 + r + 8 * half) * OUT_F + wave * HEAD_D + l] = c[r];
    }
}

// ---------------------------------------------------------------------------
// Kernel 2: per-node attention logits  attn_{s,d}[n,h] = <h[n,h,:], a_{src,dst}[h,:]>
// ---------------------------------------------------------------------------
__global__ void gat_attn(const float* __restrict__ h,
                         const float* __restrict__ a_src,
                         const float* __restrict__ a_dst,
                         float* __restrict__ as_out,
                         float* __restrict__ ad_out) {
    int tid = blockIdx.x * blockDim.x + threadIdx.x;
    if (tid >= N_NODES * N_HEADS) return;
    int n = tid >> 3, hh = tid & 7;
    const float* hp  = h + n * OUT_F + hh * HEAD_D;
    const float* asp = a_src + hh * HEAD_D;
    const float* adp = a_dst + hh * HEAD_D;
    float ss = 0.f, sd = 0.f;
    #pragma unroll
    for (int d = 0; d < HEAD_D; ++d) {
        float v = hp[d];
        ss += v * asp[d];
        sd += v * adp[d];
    }
    as_out[tid] = ss;
    ad_out[tid] = sd;
}

// ---------------------------------------------------------------------------
// Kernel 3: init output to bias, zero segment-max keys and exp-sums
// ---------------------------------------------------------------------------
__global__ void gat_init(float* __restrict__ out, const float* __restrict__ bias,
                         unsigned* __restrict__ mkey, float* __restrict__ ssum) {
    int tid = blockIdx.x * blockDim.x + threadIdx.x;
    if (tid < N_NODES * OUT_F) out[tid] = bias[tid & (OUT_F - 1)];
    if (tid < N_NODES * N_HEADS) { mkey[tid] = 0u; ssum[tid] = 0.f; }
}

// ---------------------------------------------------------------------------
// Kernel 4: edge scores e[E,8] + segment max over dst (atomic key-max)
// ---------------------------------------------------------------------------
__global__ void gat_edge_logits(const int* __restrict__ ei,
                                const float* __restrict__ ew,
                                const float* __restrict__ as_,
                                const float* __restrict__ ad_,
                                const float* __restrict__ ep_w,
                                const float* __restrict__ ep_b,
                                float* __restrict__ ebuf,
                                unsigned* __restrict__ mkey) {
    int e = blockIdx.x * blockDim.x + threadIdx.x;
    if (e >= N_EDGES) return;
    int s = ei[e], d = ei[N_EDGES + e];
    float w = ew[e];
    #pragma unroll
    for (int hh = 0; hh < N_HEADS; ++hh) {
        float v = as_[s * N_HEADS + hh] + ad_[d * N_HEADS + hh];
        v = (v > 0.f) ? v : LRELU_A * v;           // leaky_relu BEFORE edge bias
        v += w * ep_w[hh] + ep_b[hh];
        ebuf[e * N_HEADS + hh] = v;
        atomicMax(&mkey[d * N_HEADS + hh], fkey(v));
    }
}

// ---------------------------------------------------------------------------
// Kernel 5: ex = exp(e - max); segment sum over dst; store ex in-place
// ---------------------------------------------------------------------------
__global__ void gat_edge_exp(const int* __restrict__ ei,
                             float* __restrict__ ebuf,
                             const unsigned* __restrict__ mkey,
                             float* __restrict__ ssum) {
    int tid = blockIdx.x * blockDim.x + threadIdx.x;
    if (tid >= N_EDGES * N_HEADS) return;
    int e = tid >> 3, hh = tid & 7;
    int d = ei[N_EDGES + e];
    float m  = fdecode(mkey[d * N_HEADS + hh]);
    float ex = __expf(ebuf[tid] - m);
    ebuf[tid] = ex;
    atomicAdd(&ssum[d * N_HEADS + hh], ex);
}

// ---------------------------------------------------------------------------
// Kernel 6: out[dst] += h[src] * (ex / (sum+eps)); one float4 per thread
// ---------------------------------------------------------------------------
__global__ void gat_scatter(const int* __restrict__ ei,
                            const float* __restrict__ h,
                            const float* __restrict__ ebuf,
                            const float* __restrict__ ssum,
                            float* __restrict__ out) {
    long tid = (long)blockIdx.x * blockDim.x + threadIdx.x;
    if (tid >= (long)N_EDGES * 32) return;
    int e  = (int)(tid >> 5);
    int q  = (int)(tid & 31);   // float4 index in row, col = 4q
    int hh = q >> 2;            // (4q)>>4
    int s = ei[e], d = ei[N_EDGES + e];
    float attn = ebuf[e * N_HEADS + hh] / (ssum[d * N_HEADS + hh] + 1e-10f);
    const float4 hv = ((const float4*)(h + (size_t)s * OUT_F))[q];
    float* op = out + (size_t)d * OUT_F + q * 4;
    atomicAdd(op + 0, hv.x * attn);
    atomicAdd(op + 1, hv.y * attn);
    atomicAdd(op + 2, hv.z * attn);
    atomicAdd(op + 3, hv.w * attn);
}

// ---------------------------------------------------------------------------
extern "C" void kernel_launch(void* const* d_in, const int* in_sizes, int n_in,
                              void* d_out, int out_size, void* d_ws, size_t ws_size,
                              hipStream_t stream) {
    const float* x     = (const float*)d_in[0];
    const int*   ei    = (const int*)  d_in[1];
    const float* ew    = (const float*)d_in[2];
    const float* W     = (const float*)d_in[3];
    const float* a_src = (const float*)d_in[4];
    const float* a_dst = (const float*)d_in[5];
    const float* ep_w  = (const float*)d_in[6];
    const float* ep_b  = (const float*)d_in[7];
    const float* bias  = (const float*)d_in[8];
    float* out = (float*)d_out;

    float* ws    = (float*)d_ws;
    float* h     = ws + WS_H;
    float* as_   = ws + WS_AS;
    float* ad_   = ws + WS_AD;
    unsigned* mk = (unsigned*)(ws + WS_MKEY);
    float* ssum  = ws + WS_SSUM;
    float* ebuf  = ws + WS_EBUF;

    gat_init<<<(N_NODES * OUT_F + 255) / 256, 256, 0, stream>>>(out, bias, mk, ssum);
    gat_gemm_wmma<<<1024, 256, 0, stream>>>(x, W, h);
    gat_attn<<<(N_NODES * N_HEADS + 255) / 256, 256, 0, stream>>>(h, a_src, a_dst, as_, ad_);
    gat_edge_logits<<<(N_EDGES + 255) / 256, 256, 0, stream>>>(ei, ew, as_, ad_, ep_w, ep_b, ebuf, mk);
    gat_edge_exp<<<(N_EDGES * N_HEADS + 255) / 256, 256, 0, stream>>>(ei, ebuf, mk, ssum);
    gat_scatter<<<(int)(((long)N_EDGES * 32 + 255) / 256), 256, 0, stream>>>(ei, h, ebuf, ssum, out);
}